// cRPEncoding_23063974379830
// MI455X (gfx1250) — compile-verified
//
#include <hip/hip_runtime.h>
#include <math.h>
#include <stdint.h>

// ---- problem constants (must match reference) ----
#define NPTS    256
#define NPAIR   (NPTS * NPTS)          // 65536
#define LEVELS  16
#define TABLE   4096
#define FEAT    2
#define NENC    24                     // 3 * NUM_HEADS
// ---- tiling ----
#define EPG     4                      // encodings staged per block
#define NGROUP  (NENC / EPG)           // 6
#define THREADS 256                    // 8 wave32
#define PPT     2                      // pairs per thread
#define PAIRS_PER_BLOCK (THREADS * PPT) // 512
#define SLICE_FLOATS (TABLE * FEAT)    // 8192 floats
#define SLICE_BYTES  (SLICE_FLOATS * 4) // 32768 B

struct ResArr { float r[LEVELS]; };

// Issue async global->LDS copies of EPG table slices (one level) into one
// LDS buffer. 4 slices * 8 b128 ops/thread = 32 outstanding async ops (<63).
__device__ __forceinline__ void issue_slice_loads(const float* __restrict__ tables,
                                                  int e0, int l,
                                                  unsigned lds_base_addr, int tid) {
#pragma unroll
  for (int j = 0; j < EPG; ++j) {
    const float* gbase = tables + ((size_t)(e0 + j) * LEVELS + l) * SLICE_FLOATS;
#pragma unroll
    for (int it = 0; it < (SLICE_BYTES / (THREADS * 16)); ++it) {  // 8 iters
      unsigned off   = (unsigned)((tid + it * THREADS) * 16);
      unsigned laddr = lds_base_addr + (unsigned)(j * SLICE_BYTES) + off;
      // GVS mode: LDS byte addr in VGPR, 32-bit offset in VGPR, 64-bit base in SGPR pair.
      asm volatile("global_load_async_to_lds_b128 %0, %1, %2"
                   :: "v"(laddr), "v"(off), "s"(gbase)
                   : "memory");
    }
  }
}

__device__ __forceinline__ void wait_async_all() {
  asm volatile("s_wait_asynccnt 0x0" ::: "memory");
}

__global__ __launch_bounds__(THREADS, 1)
void cRPEncoding_hashgrid_kernel(const float* __restrict__ xyz,
                                 const float* __restrict__ tables,
                                 float* __restrict__ out,
                                 ResArr res) {
  // double-buffered staging of 4 slices: 2 * 4 * 4096 * 8B = 256 KB LDS
  __shared__ float2 lds[2][EPG][TABLE];

  const int tid      = threadIdx.x;
  const int e0       = blockIdx.y * EPG;
  const int pairBase = blockIdx.x * PAIRS_PER_BLOCK;

  // Per-thread relative offsets (xyz is tiny; read straight through L2).
  float rx[PPT], ry[PPT], rz[PPT];
#pragma unroll
  for (int p = 0; p < PPT; ++p) {
    int pair = pairBase + p * THREADS + tid;
    int i = pair >> 8, j = pair & (NPTS - 1);
    rx[p] = xyz[i * 3 + 0] - xyz[j * 3 + 0];
    ry[p] = xyz[i * 3 + 1] - xyz[j * 3 + 1];
    rz[p] = xyz[i * 3 + 2] - xyz[j * 3 + 2];
  }

  const unsigned lds0 = (unsigned)(uintptr_t)(&lds[0][0][0]);
  const unsigned lds1 = (unsigned)(uintptr_t)(&lds[1][0][0]);

  // Prologue: start level 0 into buffer 0.
  issue_slice_loads(tables, e0, 0, lds0, tid);

  for (int l = 0; l < LEVELS; ++l) {
    wait_async_all();       // this thread's copies for level l are in LDS
    __syncthreads();        // everyone's copies visible; prev level fully consumed

    // Overlap: kick off next level into the other buffer.
    if (l + 1 < LEVELS)
      issue_slice_loads(tables, e0, l + 1, ((l + 1) & 1) ? lds1 : lds0, tid);

    const int   b     = l & 1;
    const float res_l = res.r[l];

#pragma unroll
    for (int p = 0; p < PPT; ++p) {
      float sx = rx[p] * res_l, sy = ry[p] * res_l, sz = rz[p] * res_l;
      float bx = floorf(sx),    by = floorf(sy),    bz = floorf(sz);
      float wx = sx - bx,       wy = sy - by,       wz = sz - bz;
      int   cx = (int)bx,       cy = (int)by,       cz = (int)bz;

      // hash partials per dim (PRIMES = {1, 2654435761, 805459861}, uint32 wrap)
      unsigned hx[2] = { (unsigned)cx,                    (unsigned)(cx + 1) };
      unsigned hy[2] = { (unsigned)cy * 2654435761u,      (unsigned)(cy + 1) * 2654435761u };
      unsigned hz[2] = { (unsigned)cz * 805459861u,       (unsigned)(cz + 1) * 805459861u };
      float    wxs[2] = { 1.0f - wx, wx };
      float    wys[2] = { 1.0f - wy, wy };
      float    wzs[2] = { 1.0f - wz, wz };

      float2 acc[EPG];
#pragma unroll
      for (int j = 0; j < EPG; ++j) acc[j] = make_float2(0.0f, 0.0f);

#pragma unroll
      for (int corner = 0; corner < 8; ++corner) {
        const int o0 = (corner >> 2) & 1;   // dim 0 (reference corner order)
        const int o1 = (corner >> 1) & 1;   // dim 1
        const int o2 =  corner       & 1;   // dim 2 (fastest)
        unsigned idx = (hx[o0] ^ hy[o1] ^ hz[o2]) & (TABLE - 1);
        float    w   = (wxs[o0] * wys[o1]) * wzs[o2];
#pragma unroll
        for (int j = 0; j < EPG; ++j) {
          float2 t = lds[b][j][idx];        // ds_load_b64 gather
          acc[j].x = fmaf(w, t.x, acc[j].x);
          acc[j].y = fmaf(w, t.y, acc[j].y);
        }
      }

      const int pair = pairBase + p * THREADS + tid;
#pragma unroll
      for (int j = 0; j < EPG; ++j) {
        // out[e][pair][l*2 + f], flat over (24, 65536, 32)
        size_t o = ((size_t)(e0 + j) * NPAIR + (size_t)pair) * (LEVELS * FEAT)
                 + (size_t)(l * 2);
        *reinterpret_cast<float2*>(out + o) = acc[j];
      }
    }

    __syncthreads();  // buffer b fully consumed before it is refilled at l+2
  }
}

extern "C" void kernel_launch(void* const* d_in, const int* in_sizes, int n_in,
                              void* d_out, int out_size, void* d_ws, size_t ws_size,
                              hipStream_t stream) {
  (void)in_sizes; (void)n_in; (void)out_size; (void)d_ws; (void)ws_size;
  const float* xyz    = (const float*)d_in[0];
  const float* tables = (const float*)d_in[1];
  float*       out    = (float*)d_out;

  // RES computed exactly like numpy: b = 20**(1/15) in f64; res[l] = floor(b**l)
  ResArr res;
  const double b = pow(20.0, 1.0 / 15.0);
  for (int l = 0; l < LEVELS; ++l)
    res.r[l] = (float)floor(pow(b, (double)l));

  dim3 grid(NPAIR / PAIRS_PER_BLOCK, NGROUP, 1);  // 128 x 6
  cRPEncoding_hashgrid_kernel<<<grid, THREADS, 0, stream>>>(xyz, tables, out, res);
}